// WaveDecoder_15418932592842
// MI455X (gfx1250) — compile-verified
//
#include <hip/hip_runtime.h>
#include <hip/hip_bf16.h>

// ---------------------------------------------------------------------------
// WaveNet decode step for MI455X (gfx1250, wave32).
// Memory-bound: ~2 GB queue copy dominates (~86 us @ 23.3 TB/s). The copy is
// streamed with non-temporal hints so it does not churn the 192 MB L2 that
// the compute waves reuse for weights. Matrix math uses
// V_WMMA_F32_16X16X4_F32 (fp32-exact CDNA5 matrix path).
// ---------------------------------------------------------------------------

typedef __attribute__((ext_vector_type(2))) float v2f;
typedef __attribute__((ext_vector_type(4))) float v4f;
typedef __attribute__((ext_vector_type(8))) float v8f;

#define WMMA_F32_16x16x4(a, b, c) \
    __builtin_amdgcn_wmma_f32_16x16x4_f32(false, (a), false, (b), (short)0, (c), false, false)

__device__ __forceinline__ float sigf(float v) { return 1.0f / (1.0f + __expf(-v)); }
__device__ __forceinline__ float tanhfast(float v) { return 2.0f * sigf(2.0f * v) - 1.0f; }

// ---------------------------------------------------------------------------
// Prep: reshuffle conv_w and W1 into per-lane A-operand order for WMMA.
//  Wg : [24][8 mtile][16 kchunk][32 lane][2 j]   (K = concat(prev_k, h_k), 64)
//  W1r: [8 mtile][192 kchunk][32 lane][2 j]
// A-operand layout (f32 16x16x4): value(lane, j) = W[o = 16m + (lane&15)]
//                                                  [k = 4s + j + 2*(lane>>4)]
// ---------------------------------------------------------------------------
__global__ void wd_prep(const float* __restrict__ conv_w, const float* __restrict__ W1,
                        float* __restrict__ Wg, float* __restrict__ W1r) {
    int e = blockIdx.x * blockDim.x + threadIdx.x;
    if (e < 24 * 8 * 16 * 64) {
        int j = e & 1;
        int lane = (e >> 1) & 31;
        int s = (e >> 6) & 15;
        int m = (e >> 10) & 7;
        int l = e >> 13;
        int lh = lane >> 4, ln = lane & 15;
        int o = 16 * m + ln;
        int k = 4 * s + j + 2 * lh;
        float v;
        if (k < 32) v = conv_w[(((size_t)l * 128 + o) * 32 + k) * 2 + 0];        // prev tap
        else        v = conv_w[(((size_t)l * 128 + o) * 32 + (k - 32)) * 2 + 1]; // h tap
        Wg[e] = v;
    }
    if (e < 8 * 192 * 64) {
        int j = e & 1;
        int lane = (e >> 1) & 31;
        int s = (e >> 6) % 192;
        int m = (e >> 6) / 192;
        int lh = lane >> 4, ln = lane & 15;
        int o = 16 * m + ln;
        int k = 4 * s + j + 2 * lh;
        W1r[e] = W1[(size_t)o * 768 + k];
    }
}

// ---------------------------------------------------------------------------
// Bulk queue copy: out[l,b,c,0:160] = in[l,b,c,0:160]  (~2 GB of traffic).
// 16B NT loads via clang ext-vector (row starts are 640B = 16B-aligned);
// dest row stride is 161 floats (4B-aligned only) -> scalar NT stores.
// Non-temporal keeps this one-touch stream from evicting the compute
// kernel's weight working set from L2 (ISA temporal-hint NT policy).
// ---------------------------------------------------------------------------
__global__ void wd_copy(const float* __restrict__ q, float* __restrict__ qout, unsigned n4) {
    unsigned i4 = blockIdx.x * blockDim.x + threadIdx.x;
    if (i4 >= n4) return;
    unsigned row = i4 / 40u;          // 40 float4 per 160-float row
    unsigned t = (i4 - row * 40u) * 4u;
    v4f v = __builtin_nontemporal_load((const v4f*)(q + (size_t)row * 160 + t));
    float* o = qout + (size_t)row * 161 + t;
    __builtin_nontemporal_store(v[0], o + 0);
    __builtin_nontemporal_store(v[1], o + 1);
    __builtin_nontemporal_store(v[2], o + 2);
    __builtin_nontemporal_store(v[3], o + 3);
}

// ---------------------------------------------------------------------------
// Recurrent compute: one wave per 16-sample batch tile.
//   D/C layout: tile t, vgpr r, lane -> ch = 16t + r + 8*(lane>>4), batch = lane&15
//   B layout:   chunk s, vgpr j, lane -> k  = 4s + j + 2*(lane>>4), batch = lane&15
// ---------------------------------------------------------------------------
__launch_bounds__(32)
__global__ void wd_compute(const float* __restrict__ x, const float* __restrict__ feat,
                           const float* __restrict__ queues,
                           const float* __restrict__ Wh, const float* __restrict__ bh,
                           const float* __restrict__ Wc, const float* __restrict__ bc,
                           const float* __restrict__ Wg, const float* __restrict__ convb,
                           const float* __restrict__ W1r, const float* __restrict__ b1,
                           const float* __restrict__ W2, const float* __restrict__ b2,
                           float* __restrict__ out) {
    __shared__ float h_cur[32 * 16];     // current h, [ch][b]
    __shared__ float skips[768 * 16];    // all layer outputs, [32l+ch][b]
    __shared__ float red[64];            // cross-half head reduction

    const int lane = threadIdx.x;
    const int lh = lane >> 4, ln = lane & 15;
    const int b = blockIdx.x * 16 + ln;
    float* qout = out + 4096;            // new_queues region of d_out

    // ---- input dense: h0 = tanh(Wh@X+bh), c0 = tanh(Wc@X+bc) --------------
    float X[16];
    X[0] = x[b * 2 + 0];
    X[1] = x[b * 2 + 1];
#pragma unroll
    for (int k = 0; k < 14; k++) X[2 + k] = feat[b * 14 + k];

    v8f c_st[2];
#pragma unroll
    for (int t = 0; t < 2; t++) {
#pragma unroll
        for (int r = 0; r < 8; r++) {
            int ch = 16 * t + r + 8 * lh;
            float ah = bh[ch], ac = bc[ch];
#pragma unroll
            for (int k = 0; k < 16; k++) {
                ah += Wh[ch * 16 + k] * X[k];
                ac += Wc[ch * 16 + k] * X[k];
            }
            c_st[t][r] = tanhfast(ac);
            h_cur[ch * 16 + ln] = tanhfast(ah);
        }
    }
    __syncthreads();

    // ---- 24 dilated gated layers ------------------------------------------
    for (int l = 0; l < 24; l++) {
        const int d = 1 << (l & 7);
        const float* hsrc = (l == 0) ? h_cur : (skips + (size_t)(l - 1) * 512);

        // B operands: prev (global gather at t = 160-d) and h (from LDS)
        v2f pb[8], hb[8];
#pragma unroll
        for (int s = 0; s < 8; s++) {
            int ch0 = 4 * s + 2 * lh;
            size_t qi = (((size_t)l * 2048 + b) * 32 + ch0) * 160 + (160 - d);
            pb[s][0] = queues[qi];
            pb[s][1] = queues[qi + 160];
            hb[s][0] = hsrc[ch0 * 16 + ln];
            hb[s][1] = hsrc[(ch0 + 1) * 16 + ln];
        }

#pragma unroll
        for (int t = 0; t < 2; t++) {
            v8f gate[4];  // ig, cf, cg, eg for channels 16t..16t+15
#pragma unroll
            for (int g = 0; g < 4; g++) {
                const int m = 2 * g + t;  // gate g occupies output tiles {2g, 2g+1}
                const v4f* bp = (const v4f*)(convb + l * 128 + 16 * m + 8 * lh);
                v4f blo = bp[0], bhi = bp[1];
                v8f acc;
                acc[0] = blo[0]; acc[1] = blo[1]; acc[2] = blo[2]; acc[3] = blo[3];
                acc[4] = bhi[0]; acc[5] = bhi[1]; acc[6] = bhi[2]; acc[7] = bhi[3];
                const float* wp = Wg + (((size_t)l * 8 + m) * 16) * 64 + lane * 2;
#pragma unroll
                for (int s = 0; s < 16; s++) {
                    v2f a;
                    a[0] = wp[s * 64];
                    a[1] = wp[s * 64 + 1];
                    v2f bo = (s < 8) ? pb[s] : hb[s - 8];
                    acc = WMMA_F32_16x16x4(a, bo, acc);
                }
                gate[g] = acc;
            }
#pragma unroll
            for (int r = 0; r < 8; r++) {
                float cn = sigf(gate[0][r]) * c_st[t][r] +
                           tanhfast(gate[1][r]) * sigf(gate[2][r]);
                c_st[t][r] = cn;
                float hn = sigf(gate[3][r]) * tanhfast(cn);
                int ch = 16 * t + r + 8 * lh;
                skips[((size_t)l * 32 + ch) * 16 + ln] = hn;                // skip + next-h
                qout[(((size_t)l * 2048 + b) * 32 + ch) * 161 + 160] = hn;  // queue append
            }
        }
        __syncthreads();
    }

    // ---- head: y = relu(W1@skips + b1);  h_hat = W2@y + b2 ----------------
    float hh0 = 0.f, hh1 = 0.f;
#pragma unroll 1
    for (int m = 0; m < 8; m++) {
        const v4f* bp = (const v4f*)(b1 + 16 * m + 8 * lh);
        v4f blo = bp[0], bhi = bp[1];
        v8f Y;
        Y[0] = blo[0]; Y[1] = blo[1]; Y[2] = blo[2]; Y[3] = blo[3];
        Y[4] = bhi[0]; Y[5] = bhi[1]; Y[6] = bhi[2]; Y[7] = bhi[3];
        const float* wp = W1r + ((size_t)m * 192) * 64 + lane * 2;
#pragma unroll 8
        for (int s = 0; s < 192; s++) {
            v2f a;
            a[0] = wp[s * 64];
            a[1] = wp[s * 64 + 1];
            int ch0 = 4 * s + 2 * lh;
            v2f bo;
            bo[0] = skips[ch0 * 16 + ln];
            bo[1] = skips[(ch0 + 1) * 16 + ln];
            Y = WMMA_F32_16x16x4(a, bo, Y);
        }
#pragma unroll
        for (int r = 0; r < 8; r++) {
            float y = fmaxf(Y[r], 0.f);
            int ch = 16 * m + r + 8 * lh;
            hh0 += W2[ch] * y;
            hh1 += W2[128 + ch] * y;
        }
    }
    red[lh * 32 + ln * 2 + 0] = hh0;
    red[lh * 32 + ln * 2 + 1] = hh1;
    __syncthreads();
    if (lane < 16) {
        out[b * 2 + 0] = red[ln * 2 + 0] + red[32 + ln * 2 + 0] + b2[0];
        out[b * 2 + 1] = red[ln * 2 + 1] + red[32 + ln * 2 + 1] + b2[1];
    }
}

// ---------------------------------------------------------------------------
extern "C" void kernel_launch(void* const* d_in, const int* in_sizes, int n_in,
                              void* d_out, int out_size, void* d_ws, size_t ws_size,
                              hipStream_t stream) {
    const float* x       = (const float*)d_in[0];
    const float* feat    = (const float*)d_in[1];
    const float* queues  = (const float*)d_in[2];
    const float* W_h     = (const float*)d_in[3];
    const float* b_h     = (const float*)d_in[4];
    const float* W_c     = (const float*)d_in[5];
    const float* b_c     = (const float*)d_in[6];
    const float* conv_w  = (const float*)d_in[7];
    const float* conv_b  = (const float*)d_in[8];
    const float* W1      = (const float*)d_in[9];
    const float* b1      = (const float*)d_in[10];
    const float* W2      = (const float*)d_in[11];
    const float* b2      = (const float*)d_in[12];
    float* out = (float*)d_out;

    float* Wg  = (float*)d_ws;          // 24*8*16*64 = 196608 floats
    float* W1r = Wg + 196608;           // 8*192*64   =  98304 floats

    wd_prep<<<(196608 + 255) / 256, 256, 0, stream>>>(conv_w, W1, Wg, W1r);

    const unsigned n4 = 24u * 2048u * 32u * 40u;  // float4 chunks of queue rows
    wd_copy<<<(n4 + 255) / 256, 256, 0, stream>>>(queues, out + 4096, n4);

    wd_compute<<<128, 32, 0, stream>>>(x, feat, queues, W_h, b_h, W_c, b_c,
                                       Wg, conv_b, W1r, b1, W2, b2, out);
}